// Pesudo_RenderNet_69432441307677
// MI455X (gfx1250) — compile-verified
//
#include <hip/hip_runtime.h>

typedef __attribute__((ext_vector_type(2))) float v2f;
typedef __attribute__((ext_vector_type(8))) float v8f;

#define NCAM 5
#define IMG_H 64
#define IMG_W 176
#define ND_BINS 88
#define D0_F 1.0f
#define DSTEP_F 0.5f
#define GX 200
#define GY 200
#define GZ 16
#define NCLS 10
#define IGNORE_CLS 9
#define XMIN_F (-100.0f)
#define YMIN_F (-100.0f)
#define ZMIN_F (-3.0f)
#define RANGE_X 200.0f
#define RANGE_Y 200.0f
#define RANGE_Z 16.0f
#define OCC_RATIO_F 2.5f
#define DS_F 4.0f
#define RAYS_PER_CAM (IMG_H * IMG_W)    /* 11264, multiple of 32 */
#define NRAYS (NCAM * RAYS_PER_CAM)     /* 56320, multiple of 256 */
#define NVOX (GX * GY * GZ)             /* 640000 */

__device__ inline void inv3x3(const float a[9], float o[9]) {
  float c00 = a[4]*a[8] - a[5]*a[7];
  float c01 = a[5]*a[6] - a[3]*a[8];
  float c02 = a[3]*a[7] - a[4]*a[6];
  float id  = 1.0f / (a[0]*c00 + a[1]*c01 + a[2]*c02);
  o[0] = c00 * id;
  o[1] = (a[2]*a[7] - a[1]*a[8]) * id;
  o[2] = (a[1]*a[5] - a[2]*a[4]) * id;
  o[3] = c01 * id;
  o[4] = (a[0]*a[8] - a[2]*a[6]) * id;
  o[5] = (a[2]*a[3] - a[0]*a[5]) * id;
  o[6] = c02 * id;
  o[7] = (a[1]*a[6] - a[0]*a[7]) * id;
  o[8] = (a[0]*a[4] - a[1]*a[3]) * id;
}

__device__ inline void mul3x3(const float a[9], const float b[9], float o[9]) {
  #pragma unroll
  for (int i = 0; i < 3; ++i)
    #pragma unroll
    for (int j = 0; j < 3; ++j)
      o[i*3+j] = a[i*3+0]*b[0*3+j] + a[i*3+1]*b[1*3+j] + a[i*3+2]*b[2*3+j];
}

// Kernel 0: fuse cam2ego_rot * inv(cam2img) * inv(post_rots), scaled by OCC_RATIO.
// camMat layout per camera (16 floats): [0..8] = Mo row-major, [9..11] = to.
__global__ void prep_mats_kernel(const float* __restrict__ cam2ego_rot,
                                 const float* __restrict__ cam2ego_trans,
                                 const float* __restrict__ cam2img,
                                 const float* __restrict__ post_rots,
                                 float* __restrict__ camMat) {
  int c = blockIdx.x * blockDim.x + threadIdx.x;
  if (c >= NCAM) return;
  float R[9], K[9], P[9], iK[9], iP[9], T[9], M[9];
  #pragma unroll
  for (int i = 0; i < 9; ++i) {
    R[i] = cam2ego_rot[c*9 + i];
    K[i] = cam2img[c*9 + i];
    P[i] = post_rots[c*9 + i];
  }
  inv3x3(K, iK);
  inv3x3(P, iP);
  mul3x3(iK, iP, T);
  mul3x3(R, T, M);
  float* o = camMat + c * 16;
  #pragma unroll
  for (int i = 0; i < 9; ++i) o[i] = M[i] * OCC_RATIO_F;
  #pragma unroll
  for (int i = 0; i < 3; ++i) o[9 + i] = cam2ego_trans[c*3 + i] * OCC_RATIO_F;
  o[12] = 0.f; o[13] = 0.f; o[14] = 0.f; o[15] = 0.f;
}

// Kernel 1: per-voxel argmax over class logits (softmax is monotone -> argmax of logits).
__global__ void sem_argmax_kernel(const float* __restrict__ sem,
                                  int* __restrict__ idx) {
  int v = blockIdx.x * blockDim.x + threadIdx.x;
  if (v >= NVOX) return;
  const float* p = sem + (size_t)v * NCLS;
  float best = p[0];
  int bi = 0;
  #pragma unroll
  for (int cIt = 1; cIt < NCLS; ++cIt) {
    float x = p[cIt];
    if (x > best) { best = x; bi = cIt; }   // first-max tie rule, like jnp.argmax
  }
  idx[v] = bi;
}

// Kernel 2: WMMA-batched ray transform + first-hit depth march.
// One wave = 32 rays. Two V_WMMA_F32_16X16X4_F32 ops compute q = Mo @ v for all 32.
// A (16x4): rows 0..2 and 8..10 carry Mo (so lanes 0..15 and 16..31 of D both see q).
// B (4x16): column n = (vx, vy, vz, 0) of ray rayBase+n (WMMA#1) / rayBase+16+n (WMMA#2).
__global__ void render_kernel(const float* __restrict__ post_trans,
                              const float* __restrict__ camMat,
                              const int* __restrict__ semIdx,
                              float* __restrict__ out) {
  int gtid = blockIdx.x * blockDim.x + threadIdx.x;   // == global ray id
  int lane = threadIdx.x & 31;
  int rayBase = gtid & ~31;
  int cam = gtid / RAYS_PER_CAM;                      // wave-uniform (11264 % 32 == 0)
  const float* Mo = camMat + cam * 16;
  float pt0 = post_trans[cam*3 + 0];
  float pt1 = post_trans[cam*3 + 1];
  float pt2 = post_trans[cam*3 + 2];

  // ---- A matrix (16x4 f32, striped per ISA layout: lane<16 holds K0,K1; lane>=16 holds K2,K3)
  v2f A = {0.f, 0.f};
  int m = lane & 15;
  if ((m & 7) < 3) {
    int r = m & 7;                   // rows 0..2 and 8..10 carry Mo
    if (lane < 16) { A.x = Mo[r*3 + 0]; A.y = Mo[r*3 + 1]; }
    else           { A.x = Mo[r*3 + 2]; A.y = 0.f; }
  }

  // ---- B matrices (4x16 f32): column per lane%16; lane<16 carries K0,K1; lane>=16 K2,K3
  int col = lane & 15;
  int rcA = (rayBase + col)      - cam * RAYS_PER_CAM;
  int rcB = (rayBase + 16 + col) - cam * RAYS_PER_CAM;
  float vaX = (float)(rcA % IMG_W) * DS_F + (0.5f * DS_F) - pt0;
  float vaY = (float)(rcA / IMG_W) * DS_F + (0.5f * DS_F) - pt1;
  float vbX = (float)(rcB % IMG_W) * DS_F + (0.5f * DS_F) - pt0;
  float vbY = (float)(rcB / IMG_W) * DS_F + (0.5f * DS_F) - pt1;
  float vZ  = 1.0f - pt2;                      // camera-uniform z component
  v2f B1, B2;
  if (lane < 16) { B1.x = vaX; B1.y = vaY; B2.x = vbX; B2.y = vbY; }
  else           { B1.x = vZ;  B1.y = 0.f; B2.x = vZ;  B2.y = 0.f; }

  v8f cz = {};
  v8f Dv1 = __builtin_amdgcn_wmma_f32_16x16x4_f32(
      false, A, false, B1, (short)0, cz, false, false);
  v8f Dv2 = __builtin_amdgcn_wmma_f32_16x16x4_f32(
      false, A, false, B2, (short)0, cz, false, false);

  // Lane l < 16 reads rows M=0..2 (its ray = rayBase+l) from D of WMMA#1;
  // lane l >= 16 reads rows M=8..10 (its ray = rayBase+16+(l-16) = rayBase+l) from WMMA#2.
  float qx = (lane < 16) ? Dv1[0] : Dv2[0];
  float qy = (lane < 16) ? Dv1[1] : Dv2[1];
  float qz = (lane < 16) ? Dv1[2] : Dv2[2];

  // Fold (t*OCC_RATIO - xyz_min) so the march is g = d*q + t', in-bounds iff 0<=g<=range.
  float tox = Mo[9]  - XMIN_F;
  float toy = Mo[10] - YMIN_F;
  float toz = Mo[11] - ZMIN_F;

  __builtin_prefetch(semIdx, 0, 3);            // global_prefetch_b8

  int sem = IGNORE_CLS;                        // reference default: render[0]==IGNORE
  float depth = D0_F;                          // reference default: first_idx==0 -> D0
  for (int k = 0; k < ND_BINS; ++k) {
    float d = D0_F + DSTEP_F * (float)k;
    float gx = fmaf(d, qx, tox);
    float gy = fmaf(d, qy, toy);
    float gz = fmaf(d, qz, toz);
    bool in = (gx >= 0.f) & (gx <= RANGE_X) &
              (gy >= 0.f) & (gy <= RANGE_Y) &
              (gz >= 0.f) & (gz <= RANGE_Z);
    if (in) {
      int ix = min((int)gx, GX - 1);           // gx>=0 so (int) == floor
      int iy = min((int)gy, GY - 1);
      int iz = min((int)gz, GZ - 1);
      int lin = (ix * GY + iy) * GZ + iz;
      int s = semIdx[lin];
      if (s != IGNORE_CLS) { sem = s; depth = d; break; }   // first non-ignore hit
    }
  }
  out[gtid]         = (float)sem;              // sem_view (B,N,H,W) flat
  out[NRAYS + gtid] = depth;                   // depth_map (B,N,H,W) flat
}

extern "C" void kernel_launch(void* const* d_in, const int* in_sizes, int n_in,
                              void* d_out, int out_size, void* d_ws, size_t ws_size,
                              hipStream_t stream) {
  const float* cam2ego_rot   = (const float*)d_in[0];
  const float* cam2ego_trans = (const float*)d_in[1];
  const float* cam2img       = (const float*)d_in[2];
  const float* post_rots     = (const float*)d_in[3];
  const float* post_trans    = (const float*)d_in[4];
  // d_in[5] = bda (unused by reference forward)
  const float* sem_occ       = (const float*)d_in[6];
  // d_in[7] = sem_view_map, d_in[8] = occ_score (unused)

  float* camMat = (float*)d_ws;                          // 5*16 floats
  int*   semIdx = (int*)((char*)d_ws + 1024);            // 640000 ints

  float* out = (float*)d_out;

  prep_mats_kernel<<<1, 32, 0, stream>>>(cam2ego_rot, cam2ego_trans, cam2img,
                                         post_rots, camMat);
  sem_argmax_kernel<<<(NVOX + 255) / 256, 256, 0, stream>>>(sem_occ, semIdx);
  render_kernel<<<NRAYS / 256, 256, 0, stream>>>(post_trans, camMat, semIdx, out);
}